// HierMPNDecoder_28948079575685
// MI455X (gfx1250) — compile-verified
//
#include <hip/hip_runtime.h>
#include <math.h>

// ---------------------------------------------------------------------------
// Problem constants (match reference)
// ---------------------------------------------------------------------------
#define Ld   16
#define Zd   64
#define Hd   256
#define NTq  80000
#define NCq  30000
#define NAq  80000
#define V0d  1500
#define V1d  3000

typedef __attribute__((ext_vector_type(2))) float v2f;
typedef __attribute__((ext_vector_type(8))) float v8f;
typedef __attribute__((ext_vector_type(4))) int   v4i;

#define AS3 __attribute__((address_space(3)))

#if defined(__has_builtin)
#if __has_builtin(__builtin_amdgcn_global_load_async_to_lds_b128) && \
    __has_builtin(__builtin_amdgcn_s_wait_asynccnt)
#define HAVE_ASYNC_LDS 1
#endif
#endif

#define KC      64      // K-chunk staged per iteration
#define LDSTR   68      // padded LDS row stride (floats): 272 B = 17*16, conflict-free

// ---------------------------------------------------------------------------
// fp32 WMMA GEMM:  C[N,Nout] = act( A[N,K1+K2] @ W[Nout,K1+K2]^T + b )
// A split across A1 (K1 cols) / A2 (K2 cols) implements concat([vecs, ctx]).
// K1, K2, K are multiples of KC so a chunk never straddles the boundary.
//
// Block = 256 threads = 8 waves -> 128x64 output tile.
// Per chunk: cooperatively stage A(128x64) and B(64x64) into LDS
// (async global->LDS when available), then each wave runs 16 k-steps of
// V_WMMA_F32_16X16X4_F32 into 4 accumulators (its 16x64 strip).
// Fragment layout per ISA 7.12.2:
//   A frag: lane m = lane&15, holds A[m][k0+2*half .. +1]
//   B frag: lane n = lane&15, holds W[n][k0+2*half .. +1]
//   C/D:    VGPR i -> row (8*half + i), col = lane&15
// ---------------------------------------------------------------------------
__global__ __launch_bounds__(256)
void gemm_wmma_f32(const float* __restrict__ A1, int K1,
                   const float* __restrict__ A2, int K2,
                   const float* __restrict__ W,
                   const float* __restrict__ bias,
                   float* __restrict__ C,
                   int N, int Nout, int relu)
{
    __shared__ float sA[128 * LDSTR];   // 34816 B
    __shared__ float sB[ 64 * LDSTR];   // 17408 B

    const int K    = K1 + K2;
    const int t    = threadIdx.x;
    const int lane = t & 31;
    const int wave = t >> 5;
    const int half = lane >> 4;
    const int l16  = lane & 15;

    const int row0 = blockIdx.x * 128;
    const int col0 = blockIdx.y * 64;

    v8f acc[4] = {};

    // LDS fragment base pointers for this wave (32-bit addressing, const offsets)
    const float* ap = &sA[(wave * 16 + l16) * LDSTR + 2 * half];
    const float* bp = &sB[l16 * LDSTR + 2 * half];

    for (int kb = 0; kb < K; kb += KC) {
        // chunk source region (wave-uniform; chunks never straddle K1)
        const float* Abase;
        int kloc, lda;
        if (kb < K1) { Abase = A1; kloc = kb;      lda = K1; }
        else         { Abase = A2; kloc = kb - K1; lda = K2; }

        __syncthreads();   // previous chunk fully consumed before overwrite

        // ---- stage A chunk: 128 rows x 64 floats, 8 float4 per thread ----
#pragma unroll
        for (int i = 0; i < 8; ++i) {
            const int a = t + 256 * i;        // 0..2047
            const int r = a >> 4;             // local row 0..127
            const int f = (a & 15) << 2;      // float offset 0..60
            int gr = row0 + r; if (gr >= N) gr = N - 1;
            const float* src = Abase + (size_t)gr * lda + kloc + f;
            float* dst = &sA[r * LDSTR + f];
#ifdef HAVE_ASYNC_LDS
            __builtin_amdgcn_global_load_async_to_lds_b128((v4i*)src,
                                                           (AS3 v4i*)dst, 0, 0);
#else
            *(float4*)dst = *(const float4*)src;
#endif
        }
        // ---- stage B chunk: 64 cols x 64 floats, 4 float4 per thread ----
#pragma unroll
        for (int i = 0; i < 4; ++i) {
            const int a = t + 256 * i;        // 0..1023
            const int c = a >> 4;             // local col 0..63
            const int f = (a & 15) << 2;
            int gc = col0 + c; if (gc >= Nout) gc = Nout - 1;
            const float* src = W + (size_t)gc * K + kb + f;
            float* dst = &sB[c * LDSTR + f];
#ifdef HAVE_ASYNC_LDS
            __builtin_amdgcn_global_load_async_to_lds_b128((v4i*)src,
                                                           (AS3 v4i*)dst, 0, 0);
#else
            *(float4*)dst = *(const float4*)src;
#endif
        }
#ifdef HAVE_ASYNC_LDS
        __builtin_amdgcn_s_wait_asynccnt(0);
#endif
        __syncthreads();

        // ---- compute: 16 k-steps x 4 WMMA, operands from LDS ----
#pragma unroll
        for (int kk = 0; kk < KC; kk += 4) {
            v2f a  = *(const v2f*)(ap + kk);
            v2f b0 = *(const v2f*)(bp + kk);
            v2f b1 = *(const v2f*)(bp + 16 * LDSTR + kk);
            v2f b2 = *(const v2f*)(bp + 32 * LDSTR + kk);
            v2f b3 = *(const v2f*)(bp + 48 * LDSTR + kk);
            acc[0] = __builtin_amdgcn_wmma_f32_16x16x4_f32(false, a, false, b0,
                                                           (short)0, acc[0], false, false);
            acc[1] = __builtin_amdgcn_wmma_f32_16x16x4_f32(false, a, false, b1,
                                                           (short)0, acc[1], false, false);
            acc[2] = __builtin_amdgcn_wmma_f32_16x16x4_f32(false, a, false, b2,
                                                           (short)0, acc[2], false, false);
            acc[3] = __builtin_amdgcn_wmma_f32_16x16x4_f32(false, a, false, b3,
                                                           (short)0, acc[3], false, false);
        }
    }

    // ---- epilogue: bias + optional ReLU, masked store ----
    const int mrow = row0 + wave * 16 + half * 8;
#pragma unroll
    for (int j = 0; j < 4; ++j) {
        const int n = col0 + j * 16 + l16;
        if (n < Nout) {
            const float bv = bias ? bias[n] : 0.0f;
#pragma unroll
            for (int i = 0; i < 8; ++i) {
                const int r = mrow + i;
                if (r < N) {
                    float v = acc[j][i] + bv;
                    if (relu) v = fmaxf(v, 0.0f);
                    C[(size_t)r * Nout + n] = v;
                }
            }
        }
    }
}

// ---------------------------------------------------------------------------
// Attention context: one thread per query; src (512 KB) is L2-resident.
//   s = src[batch_idx[n]] [16,64]; att = softmax_l(s . q[n]); ctx = att @ s
// ---------------------------------------------------------------------------
__global__ __launch_bounds__(256)
void attn_ctx(const float* __restrict__ src, const int* __restrict__ bidx,
              const float* __restrict__ Q, float* __restrict__ ctx, int N)
{
    const int n = blockIdx.x * blockDim.x + threadIdx.x;
    if (n >= N) return;

    const float* s = src + (size_t)bidx[n] * (Ld * Zd);
    const float* q = Q + (size_t)n * Zd;

    float4 qv[16];
#pragma unroll
    for (int z = 0; z < 16; ++z) qv[z] = ((const float4*)q)[z];

    float att[Ld];
    float mx = -3.402823466e38f;
#pragma unroll 4
    for (int l = 0; l < Ld; ++l) {
        const float4* sr = (const float4*)(s + l * Zd);
        float acc = 0.0f;
#pragma unroll
        for (int z = 0; z < 16; ++z) {
            const float4 sv = sr[z];
            acc += sv.x * qv[z].x + sv.y * qv[z].y + sv.z * qv[z].z + sv.w * qv[z].w;
        }
        att[l] = acc;
        mx = fmaxf(mx, acc);
    }

    float denom = 0.0f;
#pragma unroll
    for (int l = 0; l < Ld; ++l) { att[l] = __expf(att[l] - mx); denom += att[l]; }
    const float inv = 1.0f / denom;

    float4* co = (float4*)(ctx + (size_t)n * Zd);
#pragma unroll 4
    for (int z = 0; z < 16; ++z) {
        float4 o = make_float4(0.f, 0.f, 0.f, 0.f);
#pragma unroll
        for (int l = 0; l < Ld; ++l) {
            const float4 sv = ((const float4*)(s + l * Zd))[z];
            const float p = att[l] * inv;
            o.x += p * sv.x; o.y += p * sv.y; o.z += p * sv.z; o.w += p * sv.w;
        }
        co[z] = o;
    }
}

// topo_scores[n] = dot(H[n,0:256], W2[0,:]) + b2[0]
__global__ __launch_bounds__(256)
void topo_out(const float* __restrict__ H, const float* __restrict__ w2,
              const float* __restrict__ b2, float* __restrict__ out, int N)
{
    const int n = blockIdx.x * blockDim.x + threadIdx.x;
    if (n >= N) return;
    const float4* h = (const float4*)(H + (size_t)n * Hd);
    const float4* w = (const float4*)w2;
    float acc = 0.0f;
#pragma unroll 8
    for (int i = 0; i < Hd / 4; ++i) {
        const float4 a = h[i], b = w[i];
        acc += a.x * b.x + a.y * b.y + a.z * b.z + a.w * b.w;
    }
    out[n] = acc + b2[0];
}

// assm_scores[n] = dot(P[n,:], Cx[n,:])   (bias already folded into P)
__global__ __launch_bounds__(256)
void assm_out(const float* __restrict__ P, const float* __restrict__ Cx,
              float* __restrict__ out, int N)
{
    const int n = blockIdx.x * blockDim.x + threadIdx.x;
    if (n >= N) return;
    const float4* p = (const float4*)(P + (size_t)n * Zd);
    const float4* c = (const float4*)(Cx + (size_t)n * Zd);
    float acc = 0.0f;
#pragma unroll
    for (int i = 0; i < Zd / 4; ++i) {
        const float4 a = p[i], b = c[i];
        acc += a.x * b.x + a.y * b.y + a.z * b.z + a.w * b.w;
    }
    out[n] = acc;
}

// ---------------------------------------------------------------------------
// Host side
// ---------------------------------------------------------------------------
static inline void launch_gemm(const float* A1, int K1, const float* A2, int K2,
                               const float* W, const float* bias, float* C,
                               int N, int Nout, int relu, hipStream_t s)
{
    dim3 grid((N + 127) / 128, (Nout + 63) / 64);
    gemm_wmma_f32<<<grid, 256, 0, s>>>(A1, K1, A2, K2, W, bias, C, N, Nout, relu);
}

extern "C" void kernel_launch(void* const* d_in, const int* in_sizes, int n_in,
                              void* d_out, int out_size, void* d_ws, size_t ws_size,
                              hipStream_t stream)
{
    (void)in_sizes; (void)n_in; (void)out_size; (void)ws_size;

    const float* src_tree  = (const float*)d_in[0];   // [128,16,64]
    const float* src_graph = (const float*)d_in[1];   // [128,16,64]
    const float* topo_vecs = (const float*)d_in[2];   // [80000,256]
    const float* cls_vecs  = (const float*)d_in[3];   // [30000,256]
    const float* assm_vecs = (const float*)d_in[4];   // [80000,256]
    const int*   topo_bidx = (const int*)d_in[5];
    const int*   cls_bidx  = (const int*)d_in[6];
    const int*   assm_bidx = (const int*)d_in[7];
    const float* A_topo_w  = (const float*)d_in[8];   // [64,256]
    const float* A_topo_b  = (const float*)d_in[9];   // [64]
    const float* A_cls_w   = (const float*)d_in[10];
    const float* A_cls_b   = (const float*)d_in[11];
    const float* A_assm_w  = (const float*)d_in[12];
    const float* A_assm_b  = (const float*)d_in[13];
    const float* topo_W1   = (const float*)d_in[14];  // [256,320]
    const float* topo_b1   = (const float*)d_in[15];
    const float* topo_W2   = (const float*)d_in[16];  // [1,256]
    const float* topo_b2   = (const float*)d_in[17];
    const float* cls_W1    = (const float*)d_in[18];  // [256,320]
    const float* cls_b1    = (const float*)d_in[19];
    const float* cls_W2    = (const float*)d_in[20];  // [1500,256]
    const float* cls_b2    = (const float*)d_in[21];
    const float* icls_W1   = (const float*)d_in[22];  // [256,320]
    const float* icls_b1   = (const float*)d_in[23];
    const float* icls_W2   = (const float*)d_in[24];  // [3000,256]
    const float* icls_b2   = (const float*)d_in[25];
    const float* Wassm_w   = (const float*)d_in[26];  // [64,256]
    const float* Wassm_b   = (const float*)d_in[27];

    float* ws = (float*)d_ws;
    size_t off = 0;
    float* Qtopo = ws + off; off += (size_t)NTq * Zd;
    float* Ctopo = ws + off; off += (size_t)NTq * Zd;
    float* Qcls  = ws + off; off += (size_t)NCq * Zd;
    float* Ccls  = ws + off; off += (size_t)NCq * Zd;
    float* Qassm = ws + off; off += (size_t)NAq * Zd;
    float* Cassm = ws + off; off += (size_t)NAq * Zd;
    float* Passm = ws + off; off += (size_t)NAq * Zd;
    float* Htopo = ws + off; off += (size_t)NTq * Hd;
    float* Hcls  = ws + off; off += (size_t)NCq * Hd;
    float* Hicls = ws + off; off += (size_t)NCq * Hd;

    float* out      = (float*)d_out;
    float* out_topo = out;
    float* out_cls  = out + (size_t)NTq;
    float* out_icls = out_cls + (size_t)NCq * V0d;
    float* out_assm = out_icls + (size_t)NCq * V1d;

    // 1) attention-query projections + assm projection (WMMA GEMMs, Nout=64)
    launch_gemm(topo_vecs, Hd, topo_vecs, 0, A_topo_w, A_topo_b, Qtopo, NTq, Zd, 0, stream);
    launch_gemm(cls_vecs,  Hd, cls_vecs,  0, A_cls_w,  A_cls_b,  Qcls,  NCq, Zd, 0, stream);
    launch_gemm(assm_vecs, Hd, assm_vecs, 0, A_assm_w, A_assm_b, Qassm, NAq, Zd, 0, stream);
    launch_gemm(assm_vecs, Hd, assm_vecs, 0, Wassm_w,  Wassm_b,  Passm, NAq, Zd, 0, stream);

    // 2) attention contexts
    attn_ctx<<<dim3((NTq + 255) / 256), 256, 0, stream>>>(src_tree,  topo_bidx, Qtopo, Ctopo, NTq);
    attn_ctx<<<dim3((NCq + 255) / 256), 256, 0, stream>>>(src_tree,  cls_bidx,  Qcls,  Ccls,  NCq);
    attn_ctx<<<dim3((NAq + 255) / 256), 256, 0, stream>>>(src_graph, assm_bidx, Qassm, Cassm, NAq);

    // 3) MLP layer 1 with implicit concat([vecs, ctx]) + ReLU
    launch_gemm(topo_vecs, Hd, Ctopo, Zd, topo_W1, topo_b1, Htopo, NTq, Hd, 1, stream);
    launch_gemm(cls_vecs,  Hd, Ccls,  Zd, cls_W1,  cls_b1,  Hcls,  NCq, Hd, 1, stream);
    launch_gemm(cls_vecs,  Hd, Ccls,  Zd, icls_W1, icls_b1, Hicls, NCq, Hd, 1, stream);

    // 4) output heads
    topo_out<<<dim3((NTq + 255) / 256), 256, 0, stream>>>(Htopo, topo_W2, topo_b2, out_topo, NTq);
    launch_gemm(Hcls,  Hd, Hcls,  0, cls_W2,  cls_b2,  out_cls,  NCq, V0d, 0, stream);
    launch_gemm(Hicls, Hd, Hicls, 0, icls_W2, icls_b2, out_icls, NCq, V1d, 0, stream);
    assm_out<<<dim3((NAq + 255) / 256), 256, 0, stream>>>(Passm, Cassm, out_assm, NAq);
}